// TrainingGraphModel_10514079941575
// MI455X (gfx1250) — compile-verified
//
#include <hip/hip_runtime.h>
#include <math.h>
#include <stdint.h>

#define BB 16
#define MM 1024
#define NN 1024
#define DD 512
#define LD 1025
#define OUTB (1025 * 1025)

// log(1/2048), log(1024/2048)
#define LOG_A  (-7.62461898f)
#define LOG_AL (-0.69314718f)

#define KC 64          // K-chunk staged in LDS
#define AS_STRIDE 68   // padded row stride (floats): 16B-aligned rows, bank-spread

typedef __attribute__((ext_vector_type(2))) float v2f;
typedef __attribute__((ext_vector_type(4))) float v4f;
typedef __attribute__((ext_vector_type(8))) float v8f;

__device__ __forceinline__ float wave_sum(float s) {
#pragma unroll
    for (int o = 16; o > 0; o >>= 1) s += __shfl_xor(s, o, 32);
    return s;
}

// ---------------------------------------------------------------- norms
// one wave32 per feature row; rn = 1/||row||
__global__ __launch_bounds__(256)
void norms_k(const float* __restrict__ tf, const float* __restrict__ df,
             float* __restrict__ rnt, float* __restrict__ rnd) {
    int wid = blockIdx.x * 8 + (threadIdx.x >> 5);
    int lane = threadIdx.x & 31;
    if (wid >= BB * (MM + NN)) return;
    const float* row;
    float* dst;
    if (wid < BB * MM) { row = tf + (size_t)wid * DD; dst = rnt + wid; }
    else { int w2 = wid - BB * MM; row = df + (size_t)w2 * DD; dst = rnd + w2; }
    float s = 0.f;
    for (int j = lane; j < DD; j += 32) { float x = row[j]; s = fmaf(x, x, s); }
    s = wave_sum(s);
    if (lane == 0) *dst = 1.0f / sqrtf(s);
}

// ---------------------------------------------------------------- init u,v
__global__ void init_uv_k(float* __restrict__ uv, int n) {
    int i = blockIdx.x * 256 + threadIdx.x;
    if (i < n) uv[i] = 0.f;
}

// ---------------------------------------------------------------- GEMM + mask
// K[i][j] = (mask * cos_sim - 1) / lambda, written once into d_out.
// Block = 128 thr (4 waves): 32(M) x 256(N); wave tile 32x64.
// A panel (shared by all 4 waves) staged via GLOBAL_LOAD_ASYNC_TO_LDS_B128
// (ASYNCcnt), WMMA A fragments then come from LDS (ds_load_b64); B fragments
// stream from L2 with plain b64 loads.
__global__ __launch_bounds__(128)
void gemm_mask_k(const float* __restrict__ tf, const float* __restrict__ df,
                 const float* __restrict__ tl, const float* __restrict__ dl,
                 const float* __restrict__ rnt, const float* __restrict__ rnd,
                 const float* __restrict__ alpha, const float* __restrict__ eps,
                 float* __restrict__ out) {
    __shared__ float As[32 * AS_STRIDE];

    const int b    = blockIdx.z;
    const int lane = threadIdx.x & 31;
    const int wave = threadIdx.x >> 5;
    const int m0   = blockIdx.y * 32;
    const int n0   = blockIdx.x * 256 + wave * 64;
    const int rA   = lane & 15;
    const int kb   = (lane >> 4) << 1;

    const float* Abase = tf + ((size_t)(b * MM + m0)) * DD;
    const float* B0 = df + ((size_t)(b * NN + n0 + rA)) * DD + kb;

    v8f acc[2][4];
#pragma unroll
    for (int i = 0; i < 2; i++)
#pragma unroll
        for (int t = 0; t < 4; t++)
            acc[i][t] = (v8f){0.f, 0.f, 0.f, 0.f, 0.f, 0.f, 0.f, 0.f};

    for (int kc = 0; kc < DD; kc += KC) {
        // ---- stage A panel: 32 rows x KC floats = 512 float4, 4 per thread
#pragma unroll
        for (int q = 0; q < 4; q++) {
            int flat = q * 128 + threadIdx.x;
            int row  = flat >> 4;         // 0..31
            int c4   = (flat & 15) << 2;  // 0,4,...,60
            const float* g = Abase + (size_t)row * DD + kc + c4;
            unsigned loff = (unsigned)(uintptr_t)&As[row * AS_STRIDE + c4];
            asm volatile("global_load_async_to_lds_b128 %0, %1, off"
                         :: "v"(loff), "v"(g) : "memory");
        }
        asm volatile("s_wait_asynccnt 0x0" ::: "memory");
        __syncthreads();

        // ---- compute on the chunk
#pragma unroll 4
        for (int kk = 0; kk < KC; kk += 4) {
            v2f a0 = *(const v2f*)&As[rA * AS_STRIDE + kb + kk];
            v2f a1 = *(const v2f*)&As[(rA + 16) * AS_STRIDE + kb + kk];
#pragma unroll
            for (int t = 0; t < 4; t++) {
                v2f bt = *(const v2f*)(B0 + (size_t)(t * 16) * DD + kc + kk);
                acc[0][t] = __builtin_amdgcn_wmma_f32_16x16x4_f32(
                    false, a0, false, bt, (short)0, acc[0][t], false, false);
                acc[1][t] = __builtin_amdgcn_wmma_f32_16x16x4_f32(
                    false, a1, false, bt, (short)0, acc[1][t], false, false);
            }
        }
        __syncthreads();
    }

    const float rl = 1.0f / (expf(eps[0]) + 0.03f);
    const float2* tl2 = (const float2*)tl;
    const float2* dl2 = (const float2*)dl;
    const int rowHalf = (lane >> 4) << 3;  // lanes 16-31 hold M = e+8

#pragma unroll
    for (int i = 0; i < 2; i++) {
#pragma unroll
        for (int t = 0; t < 4; t++) {
            const int col = n0 + t * 16 + rA;
            const float rcn = rnd[b * NN + col];
            const float2 dv = dl2[b * NN + col];
#pragma unroll
            for (int e = 0; e < 8; e++) {
                const int row = m0 + i * 16 + rowHalf + e;
                float c = acc[i][t][e] * rnt[b * MM + row] * rcn;
                float2 tv = tl2[b * MM + row];
                float dx = tv.x - dv.x, dy = tv.y - dv.y;
                float d2 = dx * dx + dy * dy;
                float kval = (d2 <= 0.09f) ? (c - 1.0f) * rl : (-rl);
                out[(size_t)b * OUTB + (size_t)row * LD + col] = kval;
            }
        }
    }
}

// ---------------------------------------------------------------- dustbin row/col
__global__ void border_k(const float* __restrict__ alpha,
                         const float* __restrict__ eps, float* __restrict__ out) {
    int t = blockIdx.x * 256 + threadIdx.x;
    const int per = LD + MM;  // full bottom row + right column
    if (t >= BB * per) return;
    int b = t / per, s = t % per;
    float rl = 1.0f / (expf(eps[0]) + 0.03f);
    float kd = (alpha[0] - 1.0f) * rl;
    size_t base = (size_t)b * OUTB;
    if (s < LD) out[base + (size_t)MM * LD + s] = kd;
    else        out[base + (size_t)(s - LD) * LD + NN] = kd;
}

// ---------------------------------------------------------------- row LSE -> u
__global__ __launch_bounds__(256)
void row_pass_k(const float* __restrict__ out, const float* __restrict__ v,
                float* __restrict__ u) {
    int wid = blockIdx.x * 8 + (threadIdx.x >> 5);
    int lane = threadIdx.x & 31;
    if (wid >= BB * LD) return;
    int b = wid / LD, i = wid % LD;
    const float* rowp = out + (size_t)b * OUTB + (size_t)i * LD;
    const float* vb = v + b * LD;
    float s = 0.f;
    for (int j = lane; j < LD; j += 32) s += __expf(rowp[j] + vb[j]);
    s = wave_sum(s);
    if (lane == 0) u[b * LD + i] = ((i < MM) ? LOG_A : LOG_AL) - __logf(s);
}

// ---------------------------------------------------------------- col LSE -> v
__global__ __launch_bounds__(256)
void col_pass_k(const float* __restrict__ out, const float* __restrict__ u,
                float* __restrict__ v) {
    int j = blockIdx.x * 256 + threadIdx.x;
    int b = blockIdx.y;
    if (j >= LD) return;
    const float* base = out + (size_t)b * OUTB + j;
    const float* ub = u + b * LD;
    float s = 0.f;
    for (int i = 0; i < LD; i++) s += __expf(base[(size_t)i * LD] + ub[i]);
    v[b * LD + j] = ((j < NN) ? LOG_A : LOG_AL) - __logf(s);
}

// ---------------------------------------------------------------- P = exp(K+u+v)*2048
// b128 in / b128 out; global-linear float4 indexing keeps 16B alignment
// (total = 16*1025*1025 is divisible by 4).
__global__ __launch_bounds__(256)
void final_k(float* __restrict__ out, const float* __restrict__ u,
             const float* __restrict__ v) {
    const long long q = (long long)blockIdx.x * 256 + threadIdx.x;
    const long long total4 = (long long)BB * OUTB / 4;
    if (q >= total4) return;
    v4f x = *(const v4f*)(out + 4 * q);
#pragma unroll
    for (int e = 0; e < 4; e++) {
        long long idx = 4 * q + e;
        int b = (int)(idx / OUTB);
        int r = (int)(idx - (long long)b * OUTB);
        int i = r / LD, j = r - i * LD;
        x[e] = 2048.0f * __expf(x[e] + u[b * LD + i] + v[b * LD + j]);
    }
    *(v4f*)(out + 4 * q) = x;
}

// ----------------------------------------------------------------
extern "C" void kernel_launch(void* const* d_in, const int* in_sizes, int n_in,
                              void* d_out, int out_size, void* d_ws, size_t ws_size,
                              hipStream_t stream) {
    const float* tf    = (const float*)d_in[0];
    const float* df    = (const float*)d_in[1];
    const float* tl    = (const float*)d_in[2];
    const float* dl    = (const float*)d_in[3];
    const float* alpha = (const float*)d_in[4];
    const float* eps   = (const float*)d_in[5];
    float* out = (float*)d_out;

    float* w   = (float*)d_ws;
    float* rnt = w;                 // BB*MM
    float* rnd = rnt + BB * MM;     // BB*NN
    float* u   = rnd + BB * NN;     // BB*LD
    float* v   = u + BB * LD;       // BB*LD  (contiguous with u)

    // zero u,v (ws is poisoned)
    int nuv = 2 * BB * LD;
    init_uv_k<<<(nuv + 255) / 256, 256, 0, stream>>>(u, nuv);

    // inverse feature norms
    norms_k<<<(BB * (MM + NN) + 7) / 8, 256, 0, stream>>>(tf, df, rnt, rnd);

    // K matrix into d_out (WMMA GEMM + cosine + mask + (x-1)/lambda)
    gemm_mask_k<<<dim3(NN / 256, MM / 32, BB), 128, 0, stream>>>(
        tf, df, tl, dl, rnt, rnd, alpha, eps, out);

    // dustbin row/col
    int nb = BB * (LD + MM);
    border_k<<<(nb + 255) / 256, 256, 0, stream>>>(alpha, eps, out);

    // Sinkhorn: 8 x (row, col) + final row, via dual potentials u,v
    dim3 gr((BB * LD + 7) / 8);
    dim3 gc((LD + 255) / 256, BB);
    for (int it = 0; it < 8; ++it) {
        row_pass_k<<<gr, 256, 0, stream>>>(out, v, u);
        col_pass_k<<<gc, 256, 0, stream>>>(out, u, v);
    }
    row_pass_k<<<gr, 256, 0, stream>>>(out, v, u);

    // P = exp(K + u + v) * (m+n)
    long long total4 = (long long)BB * OUTB / 4;
    final_k<<<dim3((unsigned)((total4 + 255) / 256)), 256, 0, stream>>>(out, u, v);
}